// Attention_12867722018973
// MI455X (gfx1250) — compile-verified
//
#include <hip/hip_runtime.h>
#include <hip/hip_bf16.h>

typedef __attribute__((ext_vector_type(2))) float v2f;
typedef __attribute__((ext_vector_type(8))) float v8f;

#define HD 256
#define BB 16
#define LQ 256
#define LC 256
#define QT 8
#define NEG_SLOPE 0.01f

// ---------------------------------------------------------------------------
// Kernel 1: fused projections, 32x32 output tile per wave (4 accumulators).
//   combined rows m in [0, B*LQ)        : qp[b][q][o] = query  . Wq^T
//   combined rows m in [B*LQ, 2*B*LQ)   : cp = context . Wc^T + bias,
//                                         stored transposed as cpT[b][o][c]
// V_WMMA_F32_16X16X4_F32; K-loop over H in steps of 4; 4 wmma per step on
// independent accumulators (reuses each A/B fragment twice).
// ---------------------------------------------------------------------------
__global__ __launch_bounds__(32)
void proj_kernel(const float* __restrict__ query,
                 const float* __restrict__ context,
                 const float* __restrict__ w_weight,   // (H, 2H) row-major
                 const float* __restrict__ w_bias,     // (H,)
                 float* __restrict__ qp,               // [B*LQ, H]
                 float* __restrict__ cpT)              // [B, H(o), LC(c)]
{
    const int tiles_n = HD / 32;             // 8
    const int tile = blockIdx.x;
    const int tn = tile % tiles_n;
    const int tm = tile / tiles_n;           // 0 .. 2*B*LQ/32 - 1

    const int lane  = threadIdx.x;
    const int half  = lane >> 4;             // 0 | 1
    const int l15   = lane & 15;
    const int kbase = half * 2;

    const bool is_cp = (tm >= (BB * LQ) / 32);
    const int  m0    = tm * 32;              // combined row base
    const int  n0    = tn * 32;              // output (o) base

    const float* X    = is_cp ? (context + (size_t)(m0 - BB * LQ) * HD)
                              : (query   + (size_t)m0 * HD);
    const int    woff = is_cp ? HD : 0;

    // A fragments: lane holds rows m0+l15 / m0+16+l15, K = k0+kbase, +1
    const float* arow0 = X + (size_t)l15 * HD;
    const float* arow1 = X + (size_t)(16 + l15) * HD;
    // B fragments: B[k][n] = W[o = n_base+l15][h = k] (+woff) -> contiguous
    const float* brow0 = w_weight + (size_t)(n0 + l15) * (2 * HD) + woff;
    const float* brow1 = w_weight + (size_t)(n0 + 16 + l15) * (2 * HD) + woff;

    v8f acc00 = {}, acc01 = {}, acc10 = {}, acc11 = {};
    #pragma unroll 4
    for (int k0 = 0; k0 < HD; k0 += 4) {
        v2f a0 = *(const v2f*)(arow0 + k0 + kbase);
        v2f a1 = *(const v2f*)(arow1 + k0 + kbase);
        v2f b0 = *(const v2f*)(brow0 + k0 + kbase);
        v2f b1 = *(const v2f*)(brow1 + k0 + kbase);
        acc00 = __builtin_amdgcn_wmma_f32_16x16x4_f32(false, a0, false, b0,
                                                      (short)0, acc00, false, false);
        acc01 = __builtin_amdgcn_wmma_f32_16x16x4_f32(false, a0, false, b1,
                                                      (short)0, acc01, false, false);
        acc10 = __builtin_amdgcn_wmma_f32_16x16x4_f32(false, a1, false, b0,
                                                      (short)0, acc10, false, false);
        acc11 = __builtin_amdgcn_wmma_f32_16x16x4_f32(false, a1, false, b1,
                                                      (short)0, acc11, false, false);
    }

    // D layout per 16x16 tile: VGPR r -> M = r + 8*half, N = l15
    if (!is_cp) {
        #pragma unroll
        for (int r = 0; r < 8; ++r) {
            const int m  = m0 + r + 8 * half;        // b*LQ + q
            const int m2 = m + 16;
            qp[(size_t)m  * HD + n0 + l15]      = acc00[r];
            qp[(size_t)m  * HD + n0 + 16 + l15] = acc01[r];
            qp[(size_t)m2 * HD + n0 + l15]      = acc10[r];
            qp[(size_t)m2 * HD + n0 + 16 + l15] = acc11[r];
        }
    } else {
        const int   o0  = n0 + l15;
        const int   o1  = n0 + 16 + l15;
        const float bo0 = w_bias[o0];
        const float bo1 = w_bias[o1];
        #pragma unroll
        for (int r = 0; r < 8; ++r) {
            const int crow  = m0 + r + 8 * half - BB * LQ;  // b*LC + c
            const int crow2 = crow + 16;
            const int b  = crow  >> 8, c  = crow  & (LC - 1);
            const int b2 = crow2 >> 8, c2 = crow2 & (LC - 1);
            cpT[((size_t)b  * HD + o0) * LC + c ] = acc00[r] + bo0;
            cpT[((size_t)b  * HD + o1) * LC + c ] = acc01[r] + bo1;
            cpT[((size_t)b2 * HD + o0) * LC + c2] = acc10[r] + bo0;
            cpT[((size_t)b2 * HD + o1) * LC + c2] = acc11[r] + bo1;
        }
    }
}

// ---------------------------------------------------------------------------
// Kernel 2: score + leaky-ReLU + mask + softmax (the true hot loop).
// One block = 256 threads = one c each, QT=8 query rows per block.
//   sw*leaky(h) = wa*h + wb*max(h,0),  wa = sw*alpha, wb = sw*(1-alpha)
// -> 1 add + 1 max + 2 fma per element, no cndmask.
// cpT read coalesced over c; each cpT load reused for QT q's. qp/sw reads are
// same-address across the wave -> LDS broadcast.
// ---------------------------------------------------------------------------
__global__ __launch_bounds__(256)
void score_softmax_kernel(const float* __restrict__ qp,
                          const float* __restrict__ cpT,
                          const int*   __restrict__ mask,          // (B, LC)
                          const float* __restrict__ score_weight,  // (1, H)
                          float* __restrict__ attn)                // (B, LQ, LC)
{
    __shared__ float qs[QT][HD];
    __shared__ float swa[HD];
    __shared__ float swb[HD];
    __shared__ float red[LC];

    const int b  = blockIdx.x / (LQ / QT);
    const int q0 = (blockIdx.x % (LQ / QT)) * QT;
    const int t  = threadIdx.x;     // == c

    for (int i = t; i < QT * HD; i += LC)
        qs[i / HD][i % HD] = qp[((size_t)(b * LQ + q0 + i / HD)) * HD + (i % HD)];
    if (t < HD) {
        const float w = score_weight[t];
        swa[t] = w * NEG_SLOPE;
        swb[t] = w * (1.0f - NEG_SLOPE);
    }
    __syncthreads();

    const int c = t;
    const float* cpb = cpT + (size_t)b * HD * LC;

    float acc[QT];
    #pragma unroll
    for (int j = 0; j < QT; ++j) acc[j] = 0.0f;

    #pragma unroll 4
    for (int o = 0; o < HD; ++o) {
        if ((o & 3) == 0 && o + 8 < HD)
            __builtin_prefetch(cpb + (size_t)(o + 8) * LC + c, 0, 0);
        const float cv = cpb[(size_t)o * LC + c];
        const float wa = swa[o];
        const float wb = swb[o];
        #pragma unroll
        for (int j = 0; j < QT; ++j) {
            const float h = qs[j][o] + cv;
            acc[j] = fmaf(wa, h, fmaf(wb, fmaxf(h, 0.0f), acc[j]));
        }
    }

    const bool masked = (mask[b * LC + c] == 0);

    #pragma unroll
    for (int j = 0; j < QT; ++j) {
        const float s = masked ? -__builtin_inff() : acc[j];

        red[t] = s;
        __syncthreads();
        for (int off = LC / 2; off > 0; off >>= 1) {
            if (t < off) red[t] = fmaxf(red[t], red[t + off]);
            __syncthreads();
        }
        const float mx = red[0];
        __syncthreads();

        const float e = __expf(s - mx);
        red[t] = e;
        __syncthreads();
        for (int off = LC / 2; off > 0; off >>= 1) {
            if (t < off) red[t] += red[t + off];
            __syncthreads();
        }
        const float inv = 1.0f / red[0];
        __syncthreads();

        attn[((size_t)(b * LQ + q0 + j)) * LC + c] = e * inv;
    }
}

// ---------------------------------------------------------------------------
// Kernel 3: attn_output[b][q][h] = sum_c attn[b][q][c] * context[b][c][h]
// 32x32 (q,h) tile per wave, fp32 WMMA, 4 accumulators, K = LC.
// ---------------------------------------------------------------------------
__global__ __launch_bounds__(32)
void out_gemm_kernel(const float* __restrict__ attn,
                     const float* __restrict__ context,
                     float* __restrict__ out)              // (B, LQ, H)
{
    const int tiles_n = HD / 32;
    const int tile = blockIdx.x;
    const int tn  = tile % tiles_n;
    const int tmp = tile / tiles_n;
    const int tm  = tmp % (LQ / 32);
    const int b   = tmp / (LQ / 32);

    const int lane  = threadIdx.x;
    const int half  = lane >> 4;
    const int l15   = lane & 15;
    const int kbase = half * 2;

    const int m0 = tm * 32;
    const int n0 = tn * 32;

    const float* A  = attn    + (size_t)b * LQ * LC;   // [q][c]
    const float* Bm = context + (size_t)b * LC * HD;   // [c][h]
    const float* arow0 = A + (size_t)(m0 + l15) * LC;
    const float* arow1 = A + (size_t)(m0 + 16 + l15) * LC;

    v8f acc00 = {}, acc01 = {}, acc10 = {}, acc11 = {};
    #pragma unroll 4
    for (int k0 = 0; k0 < LC; k0 += 4) {
        const int k = k0 + kbase;
        v2f a0 = *(const v2f*)(arow0 + k);
        v2f a1 = *(const v2f*)(arow1 + k);
        v2f b0, b1;
        b0.x = Bm[(size_t)k       * HD + n0 + l15];
        b0.y = Bm[(size_t)(k + 1) * HD + n0 + l15];
        b1.x = Bm[(size_t)k       * HD + n0 + 16 + l15];
        b1.y = Bm[(size_t)(k + 1) * HD + n0 + 16 + l15];
        acc00 = __builtin_amdgcn_wmma_f32_16x16x4_f32(false, a0, false, b0,
                                                      (short)0, acc00, false, false);
        acc01 = __builtin_amdgcn_wmma_f32_16x16x4_f32(false, a0, false, b1,
                                                      (short)0, acc01, false, false);
        acc10 = __builtin_amdgcn_wmma_f32_16x16x4_f32(false, a1, false, b0,
                                                      (short)0, acc10, false, false);
        acc11 = __builtin_amdgcn_wmma_f32_16x16x4_f32(false, a1, false, b1,
                                                      (short)0, acc11, false, false);
    }

    #pragma unroll
    for (int r = 0; r < 8; ++r) {
        const int q  = m0 + r + 8 * half;
        const int q2 = q + 16;
        out[((size_t)(b * LQ + q )) * HD + n0 + l15]      = acc00[r];
        out[((size_t)(b * LQ + q )) * HD + n0 + 16 + l15] = acc01[r];
        out[((size_t)(b * LQ + q2)) * HD + n0 + l15]      = acc10[r];
        out[((size_t)(b * LQ + q2)) * HD + n0 + 16 + l15] = acc11[r];
    }
}

// ---------------------------------------------------------------------------
extern "C" void kernel_launch(void* const* d_in, const int* in_sizes, int n_in,
                              void* d_out, int out_size, void* d_ws, size_t ws_size,
                              hipStream_t stream) {
    const float* query        = (const float*)d_in[0];
    const float* context      = (const float*)d_in[1];
    const int*   mask         = (const int*)  d_in[2];
    const float* w_weight     = (const float*)d_in[3];
    const float* w_bias       = (const float*)d_in[4];
    const float* score_weight = (const float*)d_in[5];

    float* out_attn_output = (float*)d_out;                        // (B,LQ,H)
    float* out_attn        = (float*)d_out + (size_t)BB * LQ * HD; // (B,LQ,LC)

    float* qp_ws  = (float*)d_ws;                                  // 4 MB
    float* cpT_ws = qp_ws + (size_t)BB * LQ * HD;                  // 4 MB

    // 1) projections: (2*B*LQ)/32 row-tiles x H/32 col-tiles, one wave each
    {
        const int tiles = ((BB * LQ + BB * LC) / 32) * (HD / 32);  // 2048
        proj_kernel<<<tiles, 32, 0, stream>>>(query, context, w_weight, w_bias,
                                              qp_ws, cpT_ws);
    }
    // 2) score + softmax -> attn
    {
        const int blocks = BB * (LQ / QT);                         // 512
        score_softmax_kernel<<<blocks, LC, 0, stream>>>(qp_ws, cpT_ws, mask,
                                                        score_weight, out_attn);
    }
    // 3) attn @ context -> attn_output
    {
        const int tiles = BB * (LQ / 32) * (HD / 32);              // 1024
        out_gemm_kernel<<<tiles, 32, 0, stream>>>(out_attn, context,
                                                  out_attn_output);
    }
}